// CorGCN_30416958390558
// MI455X (gfx1250) — compile-verified
//
#include <hip/hip_runtime.h>
#include <math.h>

typedef __attribute__((ext_vector_type(2))) float v2f;
typedef __attribute__((ext_vector_type(8))) float v8f;

#define Nn 50000
#define Cc 10
#define Ll 2
#define Ee 800000
#define Dd 128

// ---------------------------------------------------------------------------
// GEMM: out[M,128] = A[M,128] @ W[128,128] (+bias) (+relu), M % 16 == 0.
// 256 threads = 8 waves. Workgroup handles a 16-row x 128-col strip.
// Wave w computes columns [16w, 16w+16) using V_WMMA_F32_16X16X4_F32.
// ---------------------------------------------------------------------------
__global__ __launch_bounds__(256)
void gemm128_wmma_kernel(const float* __restrict__ A, const float* __restrict__ W,
                         const float* __restrict__ bias, float* __restrict__ out,
                         int M, int relu)
{
    __shared__ float lA[16 * 132];          // padded stride: banks (4r+k)%64 distinct
    const int row0 = blockIdx.x * 16;
    const int t = threadIdx.x;

    // cooperative load of the 16x128 A strip
    for (int idx = t; idx < 16 * 128; idx += 256) {
        int r = idx >> 7, cc = idx & 127;
        lA[r * 132 + cc] = A[(size_t)(row0 + r) * 128 + cc];
    }
    __syncthreads();

    const int wave = t >> 5;                // 0..7
    const int lane = t & 31;
    const int lh   = lane >> 4;             // lane half
    const int lr   = lane & 15;
    const int cb   = wave * 16;             // column base

    v8f acc = {};
#pragma unroll
    for (int k0 = 0; k0 < 128; k0 += 4) {
        const int ks = k0 + 2 * lh;
        v2f a, b;
        // A frag (16x4): lanes 0-15 -> K=k0,k0+1 ; lanes 16-31 -> K=k0+2,k0+3
        a.x = lA[lr * 132 + ks];
        a.y = lA[lr * 132 + ks + 1];
        // B frag (4x16): same K split, N=lr across lanes
        b.x = W[(size_t)ks * 128 + cb + lr];
        b.y = W[(size_t)(ks + 1) * 128 + cb + lr];
        acc = __builtin_amdgcn_wmma_f32_16x16x4_f32(
                  false, a, false, b, (short)0, acc, false, false);
    }

    const float bc = bias ? bias[cb + lr] : 0.0f;
#pragma unroll
    for (int r = 0; r < 8; r++) {
        // D layout: VGPR r: lanes 0-15 -> M=r ; lanes 16-31 -> M=r+8
        int m = row0 + r + 8 * lh;
        float v = acc[r] + bc;
        if (relu) v = fmaxf(v, 0.0f);
        out[(size_t)m * 128 + cb + lr] = v;
    }
}

// ---------------------------------------------------------------------------
// Small helpers for the GCN path
// ---------------------------------------------------------------------------
__global__ void fill_f32_kernel(float* __restrict__ p, float v, int n)
{
    int i = blockIdx.x * blockDim.x + threadIdx.x;
    if (i < n) p[i] = v;
}

__global__ void deg_count_kernel(const int* __restrict__ dst, float* __restrict__ deg, int E)
{
    int e = blockIdx.x * blockDim.x + threadIdx.x;
    if (e < E) atomicAdd(&deg[dst[e]], 1.0f);
}

__global__ void rsqrt_kernel(float* __restrict__ deg, int n)
{
    int i = blockIdx.x * blockDim.x + threadIdx.x;
    if (i < n) deg[i] = rsqrtf(deg[i]);     // deg >= 1 (self loops)
}

// agg[n][d] = xw[n][d] * dinv[n]^2 + b[d]   (self loop + bias)
__global__ void agg_init_kernel(const float* __restrict__ xw, const float* __restrict__ dinv,
                                const float* __restrict__ b, float* __restrict__ agg, int n)
{
    int gid = blockIdx.x * blockDim.x + threadIdx.x;
    if (gid >= n * 128) return;
    int node = gid >> 7, d = gid & 127;
    float di = dinv[node];
    agg[gid] = xw[gid] * (di * di) + b[d];
}

// 32 lanes per edge, 4 floats per lane: agg[dst] += xw[src] * dinv[src]*dinv[dst]
__global__ __launch_bounds__(256)
void edge_scatter_kernel(const int* __restrict__ src, const int* __restrict__ dst,
                         const float* __restrict__ dinv, const float* __restrict__ xw,
                         float* __restrict__ agg, int E)
{
    int gid = blockIdx.x * blockDim.x + threadIdx.x;
    int e = gid >> 5;
    if (e >= E) return;
    int l4 = (gid & 31) * 4;
    int s = src[e], d = dst[e];
    float norm = dinv[s] * dinv[d];
    const float4 xv = *(const float4*)(xw + (size_t)s * 128 + l4);
    float* ar = agg + (size_t)d * 128 + l4;
    atomicAdd(ar + 0, xv.x * norm);
    atomicAdd(ar + 1, xv.y * norm);
    atomicAdd(ar + 2, xv.z * norm);
    atomicAdd(ar + 3, xv.w * norm);
}

// ---------------------------------------------------------------------------
// One-time: q = label@Wq+bq ; Aq = q@Wk^T ; cK[c] = q[c].bk   (single block)
// ---------------------------------------------------------------------------
__global__ __launch_bounds__(128)
void prep_q_kernel(const float* __restrict__ label, const float* __restrict__ Wq,
                   const float* __restrict__ bq, const float* __restrict__ Wk,
                   const float* __restrict__ bk, float* __restrict__ Aq,
                   float* __restrict__ cK)
{
    __shared__ float q[Cc][132];
    int t = threadIdx.x;
    for (int c = 0; c < Cc; c++) {
        float s = bq[t];
        for (int g = 0; g < 128; g++) s += label[c * 128 + g] * Wq[g * 128 + t];
        q[c][t] = s;
    }
    __syncthreads();
    for (int c = 0; c < Cc; c++) {
        float s = 0.0f;
        for (int h = 0; h < 128; h++) s += q[c][h] * Wk[t * 128 + h];
        Aq[c * 128 + t] = s;
    }
    if (t < Cc) {
        float s = 0.0f;
        for (int h = 0; h < 128; h++) s += q[t][h] * bk[h];
        cK[t] = s;
    }
}

// ---------------------------------------------------------------------------
// Per-node attention score + mix, in place on gfeA [C,N,D].
// mixed[c] = softmax_j(scale*(feat[j].Aq[c] + cK[c])) @ feat    (Wv applied later)
// ---------------------------------------------------------------------------
__global__ __launch_bounds__(128)
void attn_mix_kernel(float* __restrict__ gfeA, const float* __restrict__ Aq,
                     const float* __restrict__ cK, float scale)
{
    __shared__ float feat[Cc][132];
    __shared__ float sAq[Cc][132];
    __shared__ float logit[Cc][12];
    __shared__ float P[Cc][12];
    const int n = blockIdx.x;
    const int t = threadIdx.x;          // 0..127

    for (int c = 0; c < Cc; c++) {
        feat[c][t] = gfeA[(size_t)c * Nn * 128 + (size_t)n * 128 + t];
        sAq[c][t]  = Aq[c * 128 + t];
    }
    __syncthreads();

    if (t < Cc * Cc) {
        int c = t / Cc, j = t % Cc;
        float s = 0.0f;
        for (int h = 0; h < 128; h++) s += sAq[c][h] * feat[j][h];
        logit[c][j] = (s + cK[c]) * scale;
    }
    __syncthreads();

    if (t < Cc) {
        float m = -1e30f;
        for (int j = 0; j < Cc; j++) m = fmaxf(m, logit[t][j]);
        float e[Cc], sum = 0.0f;
        for (int j = 0; j < Cc; j++) { e[j] = __expf(logit[t][j] - m); sum += e[j]; }
        float inv = 1.0f / sum;
        for (int j = 0; j < Cc; j++) P[t][j] = e[j] * inv;
    }
    __syncthreads();

    float outv[Cc];
#pragma unroll
    for (int c = 0; c < Cc; c++) {
        float s = 0.0f;
        for (int j = 0; j < Cc; j++) s += P[c][j] * feat[j][t];
        outv[c] = s;
    }
#pragma unroll
    for (int c = 0; c < Cc; c++)
        gfeA[(size_t)c * Nn * 128 + (size_t)n * 128 + t] = outv[c];
}

// ---------------------------------------------------------------------------
// ofe relative-index permutation
// ---------------------------------------------------------------------------
__global__ void board_kernel(const int* __restrict__ src_ids, int* __restrict__ board, int n)
{
    int j = blockIdx.x * blockDim.x + threadIdx.x;
    if (j < n) board[src_ids[j]] = j;
}

__global__ void permute_relu_kernel(const float* __restrict__ in, const int* __restrict__ board,
                                    const int* __restrict__ dst_ids, float* __restrict__ out,
                                    int n, int relu)
{
    int gid = blockIdx.x * blockDim.x + threadIdx.x;
    if (gid >= n * 128) return;
    int node = gid >> 7, d = gid & 127;
    int srow = board[dst_ids[node]];
    float v = in[(size_t)srow * 128 + d];
    if (relu) v = fmaxf(v, 0.0f);
    out[gid] = v;
}

// ---------------------------------------------------------------------------
static inline char* ws_take(char*& p, size_t bytes)
{
    char* r = p;
    p += (bytes + 255) & ~(size_t)255;
    return r;
}

extern "C" void kernel_launch(void* const* d_in, const int* in_sizes, int n_in,
                              void* d_out, int out_size, void* d_ws, size_t ws_size,
                              hipStream_t stream)
{
    (void)in_sizes; (void)n_in; (void)out_size; (void)ws_size;

    const float* gfe0  = (const float*)d_in[0];   // [C,N,D]
    const float* ofe0  = (const float*)d_in[1];   // [N,D]
    const float* label = (const float*)d_in[2];   // [C,D]
    const float* Wl[2] = { (const float*)d_in[3], (const float*)d_in[5] };
    const float* bl[2] = { (const float*)d_in[4], (const float*)d_in[6] };
    const float* Wq = (const float*)d_in[7];
    const float* bq = (const float*)d_in[8];
    const float* Wk = (const float*)d_in[9];
    const float* bk = (const float*)d_in[10];
    const float* Wv = (const float*)d_in[11];
    const float* bv = (const float*)d_in[12];
    const int* graph_edges = (const int*)d_in[13]; // [C,L,2,E]
    const int* ori_edges   = (const int*)d_in[14]; // [L,2,E]
    const int* src_ids     = (const int*)d_in[15]; // [L,N]
    const int* dst_ids     = (const int*)d_in[16]; // [L,N]

    float* out_gfe = (float*)d_out;                       // C*N*D
    float* out_ofe = (float*)d_out + (size_t)Cc * Nn * Dd; // N*D

    char* wp = (char*)d_ws;
    float* gfeA  = (float*)ws_take(wp, (size_t)Cc * Nn * Dd * 4); // 256 MB
    float* xw    = (float*)ws_take(wp, (size_t)Nn * Dd * 4);      // 25.6 MB
    float* ofeA  = (float*)ws_take(wp, (size_t)Nn * Dd * 4);      // 25.6 MB
    float* deg   = (float*)ws_take(wp, (size_t)Nn * 4);
    float* Aq    = (float*)ws_take(wp, (size_t)Cc * Dd * 4);
    float* cK    = (float*)ws_take(wp, 64 * 4);
    int*   board = (int*)  ws_take(wp, (size_t)Nn * 4);

    const float scale = 1.0f / sqrtf((float)Dd);
    const int TPB = 256;
    const int gN   = (Nn + TPB - 1) / TPB;
    const int gE   = (Ee + TPB - 1) / TPB;
    const int gND  = (Nn * Dd) / TPB;     // exact multiple
    const int gE32 = (Ee * 32) / TPB;     // exact multiple

    prep_q_kernel<<<1, 128, 0, stream>>>(label, Wq, bq, Wk, bk, Aq, cK);

    for (int i = 0; i < Ll; i++) {
        const float* gin = (i == 0) ? gfe0 : out_gfe;

        // ---- per-class GCN aggregation into gfeA ----
        for (int c = 0; c < Cc; c++) {
            const int* es = graph_edges + (((size_t)c * Ll + i) * 2 + 0) * Ee;
            const int* ed = es + Ee;
            float* aggc = gfeA + (size_t)c * Nn * Dd;

            fill_f32_kernel<<<gN, TPB, 0, stream>>>(deg, 1.0f, Nn);
            deg_count_kernel<<<gE, TPB, 0, stream>>>(ed, deg, Ee);
            rsqrt_kernel<<<gN, TPB, 0, stream>>>(deg, Nn);
            gemm128_wmma_kernel<<<Nn / 16, TPB, 0, stream>>>(
                gin + (size_t)c * Nn * Dd, Wl[i], nullptr, xw, Nn, 0);
            agg_init_kernel<<<gND, TPB, 0, stream>>>(xw, deg, bl[i], aggc, Nn);
            edge_scatter_kernel<<<gE32, TPB, 0, stream>>>(es, ed, deg, xw, aggc, Ee);
        }

        // ---- cross-class attention (scores+mix in place, then @Wv+bv via WMMA) ----
        attn_mix_kernel<<<Nn, 128, 0, stream>>>(gfeA, Aq, cK, scale);
        gemm128_wmma_kernel<<<(Cc * Nn) / 16, TPB, 0, stream>>>(
            gfeA, Wv, bv, out_gfe, Cc * Nn, (i == 0) ? 1 : 0);

        // ---- original-graph branch ----
        const float* oin = (i == 0) ? ofe0 : out_ofe;
        const int* os = ori_edges + ((size_t)i * 2 + 0) * Ee;
        const int* od = os + Ee;

        fill_f32_kernel<<<gN, TPB, 0, stream>>>(deg, 1.0f, Nn);
        deg_count_kernel<<<gE, TPB, 0, stream>>>(od, deg, Ee);
        rsqrt_kernel<<<gN, TPB, 0, stream>>>(deg, Nn);
        gemm128_wmma_kernel<<<Nn / 16, TPB, 0, stream>>>(oin, Wl[i], nullptr, xw, Nn, 0);
        agg_init_kernel<<<gND, TPB, 0, stream>>>(xw, deg, bl[i], ofeA, Nn);
        edge_scatter_kernel<<<gE32, TPB, 0, stream>>>(os, od, deg, xw, ofeA, Ee);

        board_kernel<<<gN, TPB, 0, stream>>>(src_ids + (size_t)i * Nn, board, Nn);
        permute_relu_kernel<<<gND, TPB, 0, stream>>>(
            ofeA, board, dst_ids + (size_t)i * Nn, out_ofe, Nn, (i == 0) ? 1 : 0);
    }
}